// TransformerLayer_85323820303106
// MI455X (gfx1250) — compile-verified
//
#include <hip/hip_runtime.h>
#include <hip/hip_bf16.h>

typedef __attribute__((ext_vector_type(2))) float v2f;
typedef __attribute__((ext_vector_type(8))) float v8f;

#define SEQ    32
#define DMODEL 8192
#define DHEAD  64
#define LN_EPS 1e-5f

// ---------------------------------------------------------------------------
// Generic M=32 GEMM:  C[32,N] = A[32,K] @ B[K,N]  (+bias)(+res)(relu)
// One block = 32 output columns; 4 waves, each wave computes one 16x16 tile
// with V_WMMA_F32_16X16X4_F32, accumulating over K in steps of 4.
//
// VGPR layouts per CDNA5 ISA 7.12.2 (32-bit, wave32):
//   A 16x4 : lane l in [0,16): row M=l, v0=K0,v1=K1 ; lanes [16,32): K2,K3
//   B 4x16 : lane l in [0,16): col N=l, v0=K0,v1=K1 ; lanes [16,32): K2,K3
//   C 16x16: VGPR g holds (M=g, lanes 0-15) and (M=g+8, lanes 16-31)
// ---------------------------------------------------------------------------
template <bool BIAS, bool RELU, bool RES>
__global__ __launch_bounds__(128) void gemm32_wmma(
    const float* __restrict__ A, int K,
    const float* __restrict__ B, int N,
    const float* __restrict__ bias,
    const float* __restrict__ res,
    float* __restrict__ C)
{
    const int lane = threadIdx.x & 31;
    const int wave = threadIdx.x >> 5;
    const int mt   = wave & 1;        // M tile: 0 or 16
    const int nt   = wave >> 1;       // N tile: 0 or 16
    const int half = lane >> 4;       // half-wave selects K pair {0,1} vs {2,3}
    const int l16  = lane & 15;
    const int n0   = blockIdx.x * 32 + nt * 16;

    // A: row-major [32,K]; this lane streams K pairs from row (mt*16+l16)
    const float* Arow = A + (mt * 16 + l16) * K + 2 * half;
    // B: row-major [K,N]; this lane streams column (n0+l16), K rows 2*half{+0,+1}
    const float* Bcol = B + (2 * half) * N + (n0 + l16);

    v8f acc = {};
    for (int k = 0; k < K; k += 32) {
        if (k + 96 < K)  // stream-ahead hint on the weight matrix
            __builtin_prefetch((const void*)(Bcol + (k + 96) * N), 0, 1);
        #pragma unroll
        for (int kk = 0; kk < 32; kk += 4) {
            v2f a = *(const v2f*)(Arow + k + kk);   // 8B aligned: indices are even
            v2f b;
            b.x = Bcol[(k + kk) * N];
            b.y = Bcol[(k + kk) * N + N];
            acc = __builtin_amdgcn_wmma_f32_16x16x4_f32(
                /*neg_a=*/false, a, /*neg_b=*/false, b,
                /*c_mod=*/(short)0, acc, /*reuse_a=*/false, /*reuse_b=*/false);
        }
    }

    const int n = n0 + l16;
    float bv = 0.f;
    if constexpr (BIAS) bv = bias[n];
    #pragma unroll
    for (int g = 0; g < 8; ++g) {
        const int m = mt * 16 + half * 8 + g;
        float v = acc[g];
        if constexpr (BIAS) v += bv;
        if constexpr (RES)  v += res[m * N + n];
        if constexpr (RELU) v = fmaxf(v, 0.f);
        C[m * N + n] = v;
    }
}

// ---------------------------------------------------------------------------
// Attention core: scores = min(q@k^T, causal_tri), softmax rows, o = a @ v.
// One wave; thread i owns row i. Tiny (32x32), pure VALU.
// ---------------------------------------------------------------------------
__global__ __launch_bounds__(32) void attn32(
    const float* __restrict__ q, const float* __restrict__ k,
    const float* __restrict__ v, float* __restrict__ o)
{
    const int i = threadIdx.x;
    float s[SEQ];
    float mx = -3.4e38f;
    for (int j = 0; j < SEQ; ++j) {
        float d = 0.f;
        for (int t = 0; t < DHEAD; ++t)
            d = fmaf(q[i * DHEAD + t], k[j * DHEAD + t], d);
        const float tri = (j <= i) ? 1.0e5f : -1.0e5f;  // (tril(2)-1)*1e5, via min
        d = fminf(d, tri);
        s[j] = d;
        mx = fmaxf(mx, d);
    }
    float sum = 0.f;
    for (int j = 0; j < SEQ; ++j) { s[j] = expf(s[j] - mx); sum += s[j]; }
    const float rs = 1.f / sum;
    for (int t = 0; t < DHEAD; ++t) {
        float acc = 0.f;
        for (int j = 0; j < SEQ; ++j)
            acc = fmaf(s[j], v[j * DHEAD + t], acc);
        o[i * DHEAD + t] = acc * rs;
    }
}

// ---------------------------------------------------------------------------
// LayerNorm over last dim (8192), no affine. One block per row.
// var is the biased variance (jnp.var default), computed as E[u^2]-E[u]^2.
// ---------------------------------------------------------------------------
__global__ __launch_bounds__(256) void ln_row(
    const float* __restrict__ u, float* __restrict__ o)
{
    const int row = blockIdx.x;
    const float* ur = u + row * DMODEL;
    float s = 0.f, s2 = 0.f;
    for (int i = threadIdx.x; i < DMODEL; i += 256) {
        const float val = ur[i];
        s += val;
        s2 = fmaf(val, val, s2);
    }
    __shared__ float rs[256], rs2[256];
    rs[threadIdx.x] = s; rs2[threadIdx.x] = s2;
    __syncthreads();
    for (int off = 128; off > 0; off >>= 1) {
        if (threadIdx.x < off) {
            rs[threadIdx.x]  += rs[threadIdx.x + off];
            rs2[threadIdx.x] += rs2[threadIdx.x + off];
        }
        __syncthreads();
    }
    const float mean  = rs[0] * (1.f / DMODEL);
    const float var   = rs2[0] * (1.f / DMODEL) - mean * mean;
    const float scale = rsqrtf(var + LN_EPS);
    float* orow = o + row * DMODEL;
    for (int i = threadIdx.x; i < DMODEL; i += 256)
        orow[i] = (ur[i] - mean) * scale;
}

// ---------------------------------------------------------------------------
// Orchestration (inputs in setup_inputs() order):
//  0:x 1:wk 2:wq 3:wv 4:w1 5:b1 6:w2 7:b2 8:w3 9:b3
// ---------------------------------------------------------------------------
extern "C" void kernel_launch(void* const* d_in, const int* in_sizes, int n_in,
                              void* d_out, int out_size, void* d_ws, size_t ws_size,
                              hipStream_t stream)
{
    const float* x  = (const float*)d_in[0];
    const float* wk = (const float*)d_in[1];
    const float* wq = (const float*)d_in[2];
    const float* wv = (const float*)d_in[3];
    const float* w1 = (const float*)d_in[4];
    const float* b1 = (const float*)d_in[5];
    const float* w2 = (const float*)d_in[6];
    const float* b2 = (const float*)d_in[7];
    const float* w3 = (const float*)d_in[8];
    const float* b3 = (const float*)d_in[9];
    float* out = (float*)d_out;

    float* ws = (float*)d_ws;
    float* qb = ws;                       // [32,64]
    float* kb = qb + SEQ * DHEAD;         // [32,64]
    float* vb = kb + SEQ * DHEAD;         // [32,64]
    float* ab = vb + SEQ * DHEAD;         // [32,64] attention output
    float* tb = ab + SEQ * DHEAD;         // [32,8192] pre-LN1
    float* yb = tb + SEQ * DMODEL;        // [32,8192] post-LN1
    float* hb = yb + SEQ * DMODEL;        // [32,8192] relu(y@w2+b2)
    float* sb = hb + SEQ * DMODEL;        // [32,8192] pre-LN2

    // q,k,v projections: [32,8192]@[8192,64]
    gemm32_wmma<false, false, false><<<2, 128, 0, stream>>>(x, DMODEL, wq, DHEAD, nullptr, nullptr, qb);
    gemm32_wmma<false, false, false><<<2, 128, 0, stream>>>(x, DMODEL, wk, DHEAD, nullptr, nullptr, kb);
    gemm32_wmma<false, false, false><<<2, 128, 0, stream>>>(x, DMODEL, wv, DHEAD, nullptr, nullptr, vb);

    // masked softmax attention
    attn32<<<1, 32, 0, stream>>>(qb, kb, vb, ab);

    // t = attn@w1 + b1 + x ; y = LN(t)
    gemm32_wmma<true, false, true><<<DMODEL / 32, 128, 0, stream>>>(ab, DHEAD, w1, DMODEL, b1, x, tb);
    ln_row<<<SEQ, 256, 0, stream>>>(tb, yb);

    // h = relu(y@w2 + b2) ; s = h@w3 + b3 + y ; out = LN(s)
    gemm32_wmma<true, true,  false><<<DMODEL / 32, 128, 0, stream>>>(yb, DMODEL, w2, DMODEL, b2, nullptr, hb);
    gemm32_wmma<true, false, true ><<<DMODEL / 32, 128, 0, stream>>>(hb, DMODEL, w3, DMODEL, b3, yb, sb);
    ln_row<<<SEQ, 256, 0, stream>>>(sb, out);
}